// LocalSelfMultiheadAttention_2173253452656
// MI455X (gfx1250) — compile-verified
//
#include <hip/hip_runtime.h>
#include <hip/hip_bf16.h>
#include <math.h>

// Problem constants (from reference setup_inputs)
#define NF    4
#define NLOC  2048
#define NNEI  64
#define FD    256
#define HEADS 8
#define HD    32
#define NQKV  (3 * FD)          // 768
#define NROWS (NF * NLOC)       // 8192
#define SCALING 0.17677669529663687f   // 1/sqrt(32)

typedef __attribute__((ext_vector_type(2))) float v2f;
typedef __attribute__((ext_vector_type(8))) float v8f;

// ---------------------------------------------------------------------------
// Kernel 1: qkv = query @ W + b   (f32 WMMA 16x16x4)
//   query: (8192, 256) row-major, W: (256, 768) row-major, b: (768)
//   qkv:   (8192, 768) in workspace; q-columns (0..255) pre-scaled by hd^-0.5
// One wave computes one 16x16 tile of the output; 8 waves/block share M-rows.
// ---------------------------------------------------------------------------
__global__ __launch_bounds__(256) void qkv_gemm_wmma(
    const float* __restrict__ Q,
    const float* __restrict__ W,
    const float* __restrict__ B,
    float* __restrict__ qkv)
{
  const int lane = threadIdx.x & 31;
  const int wave = threadIdx.x >> 5;     // 0..7
  const int l15  = lane & 15;
  const int lh   = lane >> 4;            // 0 or 1 (lane half)

  const int n0 = (blockIdx.x * 8 + wave) * 16;   // output column tile
  const int m0 = blockIdx.y * 16;                // output row tile

  // C/D 16x16 f32 layout: VGPR r, lanes0-15 -> (M=r,     N=lane),
  //                               lanes16-31 -> (M=r+8, N=lane-16).
  // Bias depends only on N -> identical across all 8 accumulator regs.
  const float bias = B[n0 + l15];
  v8f c;
#pragma unroll
  for (int r = 0; r < 8; ++r) c[r] = bias;

  // A 16x4 f32 layout: lanes0-15 hold (M=lane, K=k0+0 / k0+1),
  //                    lanes16-31 hold (M=lane-16, K=k0+2 / k0+3).
  const float* arow = Q + (size_t)(m0 + l15) * FD + 2 * lh;
  // B 4x16 layout (rows striped across lanes, mirrored like C):
  //   lanes0-15: rows k0+0 / k0+1, lanes16-31: rows k0+2 / k0+3, col = n0+l15.
  const float* bcol = W + (size_t)(2 * lh) * NQKV + n0 + l15;

#pragma unroll 4
  for (int k0 = 0; k0 < FD; k0 += 4) {
    v2f a, bf;
    a.x = arow[k0 + 0];                       // fuses into one b64 load
    a.y = arow[k0 + 1];
    bf.x = bcol[(size_t)(k0 + 0) * NQKV];
    bf.y = bcol[(size_t)(k0 + 1) * NQKV];
    c = __builtin_amdgcn_wmma_f32_16x16x4_f32(
        /*neg_a=*/false, a, /*neg_b=*/false, bf,
        /*c_mod=*/(short)0, c, /*reuse_a=*/false, /*reuse_b=*/false);
  }

  // q columns get the attention scaling baked in at store time.
  const float sc = (n0 + l15 < FD) ? SCALING : 1.0f;
  float* orow = qkv + (size_t)(m0 + 8 * lh) * NQKV + n0 + l15;
#pragma unroll
  for (int r = 0; r < 8; ++r)
    orow[(size_t)r * NQKV] = c[r] * sc;
}

// ---------------------------------------------------------------------------
// Kernel 2: gathered attention. One block per (f, l). 256 threads = 8 waves.
//   Phase 1: 512 logits (2 per thread), mask -> -inf, + bias, emit aw.
//   Phase 2: per-head softmax, one wave32 per head (shfl_xor reductions).
//   Phase 3: o[h,d] = sum_n attn[h,n] * v[nlist[n], h, d]  (coalesced v reads)
// k/v rows are gathered from the qkv workspace; per-frame k/v (4 MB) lives in L2.
// ---------------------------------------------------------------------------
__global__ __launch_bounds__(256) void attn_gather_softmax(
    const float* __restrict__ qkv,        // (8192, 768), q pre-scaled
    const float* __restrict__ attn_bias,  // (4, 8, 2048, 64)
    const unsigned char* __restrict__ nlist_mask,  // (4, 2048, 64) bool
    const int* __restrict__ nlist,        // (4, 2048, 64)
    float* __restrict__ o_out,            // (4, 2048, 256)
    float* __restrict__ aw_out,           // (4, 8, 2048, 64)
    float* __restrict__ attn_out)         // (4*8*2048, 1, 64) == aw layout
{
  __shared__ float q_s[FD];
  __shared__ float aw_s[HEADS * NNEI];
  __shared__ int   nl_s[NNEI];
  __shared__ unsigned char mk_s[NNEI];

  const int t  = threadIdx.x;
  const int fl = blockIdx.x;             // f*NLOC + l
  const int f  = fl >> 11;               // / NLOC
  const int l  = fl & (NLOC - 1);

  q_s[t] = qkv[(size_t)fl * NQKV + t];   // q row (already scaled)
  if (t < NNEI) {
    nl_s[t] = nlist[(size_t)fl * NNEI + t];
    mk_s[t] = nlist_mask[(size_t)fl * NNEI + t];
  }
  __syncthreads();

  // ---- Phase 1: logits ----
  {
    const int n  = t & (NNEI - 1);       // neighbor 0..63
    const int h0 = t >> 6;               // 0..3; this thread does h0 and h0+4
    const int j  = nl_s[n];
    const bool m = mk_s[n] != 0;
    const float* krow = qkv + ((size_t)f * NLOC + j) * NQKV + FD;
#pragma unroll
    for (int hh = 0; hh < 2; ++hh) {
      const int h = h0 + 4 * hh;
      const float* kp = krow + h * HD;
      const float* qp = q_s + h * HD;
      float acc = 0.f;
#pragma unroll
      for (int d = 0; d < HD; d += 4)
        acc += qp[d] * kp[d] + qp[d + 1] * kp[d + 1]
             + qp[d + 2] * kp[d + 2] + qp[d + 3] * kp[d + 3];
      float val = m ? acc : -INFINITY;   // mask first (matches reference order)
      const size_t bidx = (((size_t)f * HEADS + h) * NLOC + l) * NNEI + n;
      val += attn_bias[bidx];            // -inf + finite stays -inf
      aw_s[h * NNEI + n] = val;
      aw_out[bidx] = val;
    }
  }
  __syncthreads();

  // ---- Phase 2: softmax over nnei, wave w owns head w ----
  {
    const int h    = t >> 5;             // 0..7 (wave id)
    const int lane = t & 31;
    float v0 = aw_s[h * NNEI + lane];
    float v1 = aw_s[h * NNEI + lane + 32];
    float mx = fmaxf(v0, v1);
#pragma unroll
    for (int off = 16; off > 0; off >>= 1)
      mx = fmaxf(mx, __shfl_xor(mx, off, 32));
    float e0 = (v0 == -INFINITY) ? 0.f : __expf(v0 - mx);
    float e1 = (v1 == -INFINITY) ? 0.f : __expf(v1 - mx);
    float s = e0 + e1;
#pragma unroll
    for (int off = 16; off > 0; off >>= 1)
      s += __shfl_xor(s, off, 32);
    const float inv = (s > 0.f) ? 1.f / s : 0.f;
    e0 *= inv;
    e1 *= inv;
    const size_t abase = (((size_t)f * HEADS + h) * NLOC + l) * NNEI;
    attn_out[abase + lane]      = e0;
    attn_out[abase + lane + 32] = e1;
    // Each lane exclusively owns slots (lane, lane+32) of its head: safe to
    // overwrite in place within the wave, no extra barrier needed here.
    aw_s[h * NNEI + lane]      = e0;
    aw_s[h * NNEI + lane + 32] = e1;
  }
  __syncthreads();

  // ---- Phase 3: o = attn @ v (gathered) ----
  {
    const int h = t >> 5;                // 0..7
    const int d = t & 31;                // 0..31; t == h*HD + d
    const float* vbase = qkv + (size_t)f * NLOC * NQKV + 2 * FD + h * HD + d;
    float acc = 0.f;
#pragma unroll 8
    for (int n = 0; n < NNEI; ++n)
      acc += aw_s[h * NNEI + n] * vbase[(size_t)nl_s[n] * NQKV];
    o_out[(size_t)fl * FD + t] = acc;
  }
}

// ---------------------------------------------------------------------------
extern "C" void kernel_launch(void* const* d_in, const int* in_sizes, int n_in,
                              void* d_out, int out_size, void* d_ws, size_t ws_size,
                              hipStream_t stream)
{
  (void)in_sizes; (void)n_in; (void)out_size; (void)ws_size;

  const float*         query = (const float*)d_in[0];
  const float*         bias  = (const float*)d_in[1];
  const unsigned char* mask  = (const unsigned char*)d_in[2];
  const int*           nlist = (const int*)d_in[3];
  const float*         W     = (const float*)d_in[4];
  const float*         b     = (const float*)d_in[5];

  float* out  = (float*)d_out;
  float* o    = out;                                        // 4*2048*256
  float* aw   = out + (size_t)NF * NLOC * FD;               // 4*8*2048*64
  float* attn = aw  + (size_t)NF * HEADS * NLOC * NNEI;     // 4*8*2048*64

  float* qkv = (float*)d_ws;                                // 8192*768 f32 = 24 MB

  // GEMM: tiles = (768/16) x (8192/16) = 48 x 512; 8 N-tiles per block.
  dim3 gridA(NQKV / (16 * 8), NROWS / 16);                  // (6, 512)
  qkv_gemm_wmma<<<gridA, 256, 0, stream>>>(query, W, b, qkv);

  attn_gather_softmax<<<NROWS, 256, 0, stream>>>(
      qkv, bias, mask, nlist, o, aw, attn);
}